// ABRLovaszCELoss_13176959664767
// MI455X (gfx1250) — compile-verified
//
#include <hip/hip_runtime.h>
#include <stdint.h>

#define IGNORE_L 255
#define NB 4
#define HS 96
#define HO 384
#define WO 384
#define HWO (HO * WO)            /* 147456 */
#define NPIX (NB * HO * WO)      /* 589824 = 2304*256 */
#define NPAD (1u << 20)          /* 1048576 */

typedef __attribute__((ext_vector_type(2))) float v2f;
typedef __attribute__((ext_vector_type(8))) float v8f;

#if __has_builtin(__builtin_amdgcn_global_load_async_to_lds_b64) && \
    __has_builtin(__builtin_amdgcn_s_wait_asynccnt)
#define USE_ASYNC_LDS 1
typedef int v2i __attribute__((ext_vector_type(2)));
typedef __attribute__((address_space(1))) v2i as1_v2i;
typedef __attribute__((address_space(3))) v2i as3_v2i;
#else
#define USE_ASYNC_LDS 0
#endif

// ---------------------------------------------------------------- utilities
__global__ void zero_stats_k(float* s) {
  if (threadIdx.x < 64) s[threadIdx.x] = 0.0f;
}

// Build bilinear (align_corners) interpolation matrix R[384x96] and Rt[96x384].
__global__ void build_interp_k(float* __restrict__ R, float* __restrict__ Rt) {
  int k = blockIdx.x * blockDim.x + threadIdx.x;
  if (k >= HO) return;
  float s = (float)k * (float)(HS - 1) / (float)(HO - 1);
  int i0 = (int)floorf(s);
  i0 = (i0 < 0) ? 0 : ((i0 > HS - 2) ? HS - 2 : i0);
  float t = s - (float)i0;
  for (int j = 0; j < HS; ++j) {
    float w = (j == i0) ? (1.0f - t) : ((j == i0 + 1) ? t : 0.0f);
    R[k * HS + j]  = w;
    Rt[j * HO + k] = w;
  }
}

// ------------------------------------------------ WMMA f32 GEMM (wave32)
// D[M,N] = A[M,K] * B[K,N], row-major f32, via V_WMMA_F32_16X16X4_F32.
// One wave computes a 32x16 output region (two 16x16 tiles sharing each
// B fragment: 2 WMMAs per B-fragment load).
// A-frag (16x4): lanes 0-15 rows, K={k0,k0+1}; lanes 16-31 K={k0+2,k0+3}.
// C/D (16x16): VGPR v -> row v (lanes 0-15) / row v+8 (lanes 16-31).
__global__ void gemm_wmma_k(const float* __restrict__ A, const float* __restrict__ Bm,
                            float* __restrict__ D, int M, int N, int K,
                            long sA, long sB, long sD) {
  int lane = threadIdx.x;            // 0..31, full wave, EXEC all-ones
  A  += (long)blockIdx.z * sA;
  Bm += (long)blockIdx.z * sB;
  D  += (long)blockIdx.z * sD;
  int row0  = blockIdx.y * 32 + (lane & 15);
  int row1  = row0 + 16;
  int col   = blockIdx.x * 16 + (lane & 15);
  int khalf = (lane >> 4) << 1;      // 0 or 2
  v8f c0 = {0.f, 0.f, 0.f, 0.f, 0.f, 0.f, 0.f, 0.f};
  v8f c1 = c0;
  for (int k0 = 0; k0 < K; k0 += 4) {
    int kb = k0 + khalf;
    v2f a0, a1, b;
    a0.x = A[(long)row0 * K + kb];
    a0.y = A[(long)row0 * K + kb + 1];
    a1.x = A[(long)row1 * K + kb];
    a1.y = A[(long)row1 * K + kb + 1];
    b.x  = Bm[(long)kb * N + col];
    b.y  = Bm[(long)(kb + 1) * N + col];
    if (k0 + 4 < K) __builtin_prefetch(&Bm[(long)(kb + 4) * N + col], 0, 1);
    c0 = __builtin_amdgcn_wmma_f32_16x16x4_f32(false, a0, false, b, (short)0, c0,
                                               false, false);
    c1 = __builtin_amdgcn_wmma_f32_16x16x4_f32(false, a1, false, b, (short)0, c1,
                                               false, false);
  }
  int rbase = blockIdx.y * 32 + ((lane >> 4) << 3);
#pragma unroll
  for (int v = 0; v < 8; ++v)
    D[(long)(rbase + v) * N + col] = c0[v];
#pragma unroll
  for (int v = 0; v < 8; ++v)
    D[(long)(rbase + 16 + v) * N + col] = c1[v];
}

// ------------------------------------------------ cross-entropy (masked mean)
template <int C>
__global__ void ce_kernel(const float* __restrict__ up, const int* __restrict__ lab,
                          float* __restrict__ acc) {
  __shared__ float rn[256], rv[256];
  int  l = threadIdx.x;
  long p = (long)blockIdx.x * 256 + l;
  float nll = 0.0f, vld = 0.0f;
  if (p < NPIX) {
    int b   = (int)(p / HWO);
    int rem = (int)(p % HWO);
    const float* base = up + ((long)b * C) * (long)HWO + rem;
    float z[C];
    float m = -3.0e38f;
#pragma unroll
    for (int c = 0; c < C; ++c) { z[c] = base[(long)c * HWO]; m = fmaxf(m, z[c]); }
    float s = 0.0f;
#pragma unroll
    for (int c = 0; c < C; ++c) s += __expf(z[c] - m);
    int L  = lab[p];
    int ok = (L != IGNORE_L);
    int lc = L < 0 ? 0 : (L > C - 1 ? C - 1 : L);
    float v = __logf(s) + m - z[lc];
    nll = ok ? v : 0.0f;
    vld = ok ? 1.0f : 0.0f;
  }
  rn[l] = nll; rv[l] = vld;
  __syncthreads();
  for (int off = 128; off; off >>= 1) {
    if (l < off) { rn[l] += rn[l + off]; rv[l] += rv[l + off]; }
    __syncthreads();
  }
  if (l == 0) { atomicAdd(&acc[0], rn[0]); atomicAdd(&acc[1], rv[0]); }
}

// ------------------------------------------------ softmax over channels, in place
template <int C>
__global__ void softmax_inplace_k(float* __restrict__ up) {
  long p = (long)blockIdx.x * 256 + threadIdx.x;
  if (p >= NPIX) return;
  int b   = (int)(p / HWO);
  int rem = (int)(p % HWO);
  float* base = up + ((long)b * C) * (long)HWO + rem;
  float z[C];
  float m = -3.0e38f;
#pragma unroll
  for (int c = 0; c < C; ++c) { z[c] = base[(long)c * HWO]; m = fmaxf(m, z[c]); }
  float s = 0.0f;
#pragma unroll
  for (int c = 0; c < C; ++c) { z[c] = __expf(z[c] - m); s += z[c]; }
  float inv = 1.0f / s;
#pragma unroll
  for (int c = 0; c < C; ++c) base[(long)c * HWO] = z[c] * inv;
}

// ------------------------------------------------ Lovasz: build sort keys
// key (hi32): float error mapped so ascending-u64 sort == descending error.
// payload (lo32): bit1 = valid, bit0 = fg. Padding (err=-2) sorts last.
__global__ void build_errors_k(const float* __restrict__ probs,
                               const int* __restrict__ lab, int C, int cls,
                               unsigned long long* __restrict__ sb,
                               float* __restrict__ gts) {
  __shared__ float rf[256];
  int  l = threadIdx.x;
  long t = (long)blockIdx.x * 256 + l;
  float err = -2.0f; int fg = 0, val = 0;
  if (t < NPIX) {
    int b   = (int)(t / HWO);
    int rem = (int)(t % HWO);
    int L   = lab[t];
    val = (L != IGNORE_L);
    fg  = (val && (L == cls)) ? 1 : 0;
    float pr = probs[((long)b * C + cls) * (long)HWO + rem];
    err = val ? fabsf((float)fg - pr) : -1.0f;
  }
  unsigned int bb  = __float_as_uint(err);
  unsigned int ua  = (bb & 0x80000000u) ? ~bb : (bb | 0x80000000u);
  unsigned int key = ~ua;
  sb[t] = ((unsigned long long)key << 32) | ((unsigned)val << 1) | (unsigned)fg;
  rf[l] = (float)fg;
  __syncthreads();
  for (int off = 128; off; off >>= 1) {
    if (l < off) rf[l] += rf[l + off];
    __syncthreads();
  }
  if (l == 0 && rf[0] != 0.0f) atomicAdd(gts, rf[0]);
}

// ------------------------------------------------ bitonic sort step (u64 asc)
__global__ void bitonic_step_k(unsigned long long* __restrict__ d,
                               unsigned int j, unsigned int k) {
  unsigned int i   = blockIdx.x * 256u + threadIdx.x;
  unsigned int ixj = i ^ j;
  if (ixj > i) {
    unsigned long long a = d[i], b = d[ixj];
    bool up = ((i & k) == 0u);
    if ((a > b) == up) { d[i] = b; d[ixj] = a; }
  }
}

// ------------------------------------------------ Lovasz finalize (1 block)
// Sequential chunked LDS scan of fg/valid over the sorted first NPIX entries,
// computing sum(err * d(Jaccard)) for one class. The next 2 KB chunk is
// prefetched into LDS with GLOBAL_LOAD_ASYNC_TO_LDS_B64 (ASYNCcnt) while the
// current chunk is scanned; async loads complete in order so waiting
// asynccnt<=1 guarantees the current buffer has landed.
__global__ void lovasz_finalize_k(const unsigned long long* __restrict__ sb,
                                  const float* __restrict__ gtsArr,
                                  float* __restrict__ pc, int gidx) {
  __shared__ unsigned long long dbuf[2][256];
  __shared__ float sfg[256], sval[256], sred[256];
  int   l   = threadIdx.x;
  float gts = gtsArr[gidx];
  float run_fg = 0.0f, run_val = 0.0f, accum = 0.0f;
  const int CH = NPIX / 256;  // 2304, exact
#if USE_ASYNC_LDS
  __builtin_amdgcn_global_load_async_to_lds_b64(
      (as1_v2i*)(sb + l), (as3_v2i*)&dbuf[0][l], 0, 0);
#else
  dbuf[0][l] = sb[l];
#endif
  for (int chunk = 0; chunk < CH; ++chunk) {
    int cur = chunk & 1, nxt = cur ^ 1;
    if (chunk + 1 < CH) {
#if USE_ASYNC_LDS
      __builtin_amdgcn_global_load_async_to_lds_b64(
          (as1_v2i*)(sb + (long)(chunk + 1) * 256 + l),
          (as3_v2i*)&dbuf[nxt][l], 0, 0);
      __builtin_amdgcn_s_wait_asynccnt(1);
#else
      dbuf[nxt][l] = sb[(long)(chunk + 1) * 256 + l];
#endif
    } else {
#if USE_ASYNC_LDS
      __builtin_amdgcn_s_wait_asynccnt(0);
#endif
    }
    unsigned long long d = dbuf[cur][l];
    unsigned int key = (unsigned int)(d >> 32);
    unsigned int ua  = ~key;
    unsigned int bb  = (ua & 0x80000000u) ? (ua & 0x7FFFFFFFu) : ~ua;
    float err = __uint_as_float(bb);
    int   fg  = (int)(d & 1u);
    int   val = (int)((d >> 1) & 1u);
    sfg[l]  = (float)fg;
    sval[l] = (float)val;
    __syncthreads();
    for (int off = 1; off < 256; off <<= 1) {
      float a1 = (l >= off) ? sfg[l - off]  : 0.0f;
      float a2 = (l >= off) ? sval[l - off] : 0.0f;
      __syncthreads();
      sfg[l] += a1; sval[l] += a2;
      __syncthreads();
    }
    float cfg  = run_fg  + sfg[l];
    float cval = run_val + sval[l];
    float pfg  = run_fg  + (l ? sfg[l - 1]  : 0.0f);
    float pval = run_val + (l ? sval[l - 1] : 0.0f);
    float jac  = 1.0f - (gts - cfg) / fmaxf(gts + cval - cfg, 1.0f);
    float jacp = 1.0f - (gts - pfg) / fmaxf(gts + pval - pfg, 1.0f);
    sred[l] = (val ? err : 0.0f) * (jac - jacp);
    __syncthreads();
    for (int off = 128; off; off >>= 1) {
      if (l < off) sred[l] += sred[l + off];
      __syncthreads();
    }
    if (l == 0) accum += sred[0];
    float lf = sfg[255], lv = sval[255];
    __syncthreads();
    run_fg  += lf;
    run_val += lv;
  }
  if (l == 0) pc[gidx] = accum;
}

// ------------------------------------------------ final scalar combine
__global__ void final_combine_k(const float* __restrict__ acc,
                                const float* __restrict__ gts,
                                const float* __restrict__ pc,
                                float* __restrict__ out) {
  if (threadIdx.x != 0 || blockIdx.x != 0) return;
  float ce0 = acc[0] / fmaxf(acc[1], 1.0f);
  float ced = acc[2] / fmaxf(acc[3], 1.0f);
  const int offs[4] = {0, 7, 10, 12};
  float lov[3];
  for (int br = 0; br < 3; ++br) {
    float s = 0.0f, n = 0.0f;
    for (int c = offs[br]; c < offs[br + 1]; ++c)
      if (gts[c] > 0.0f) { s += pc[c]; n += 1.0f; }
    lov[br] = s / fmaxf(n, 1.0f);
  }
  out[0] = ce0 + lov[0] + 0.4f * lov[1] + 0.4f * lov[2] + 0.4f * ced;
}

// ================================================================ host side
extern "C" void kernel_launch(void* const* d_in, const int* in_sizes, int n_in,
                              void* d_out, int out_size, void* d_ws, size_t ws_size,
                              hipStream_t stream) {
  const float* preds0 = (const float*)d_in[0];
  const float* preds1 = (const float*)d_in[1];
  const float* preds2 = (const float*)d_in[2];
  const float* predsd = (const float*)d_in[3];
  const int*   t0     = (const int*)d_in[4];
  const int*   t1     = (const int*)d_in[5];
  const int*   t2     = (const int*)d_in[6];

  char*  ws  = (char*)d_ws;
  size_t off = 0;
  auto carve = [&](size_t bytes) -> void* {
    void* p = ws + off;
    off += (bytes + 255) & ~((size_t)255);
    return p;
  };
  float* R     = (float*)carve((size_t)HO * HS * 4);        // 147456 B
  float* Rt    = (float*)carve((size_t)HS * HO * 4);        // 147456 B
  float* stats = (float*)carve(256);   // [0..3]=acc, [8..23]=gts, [32..47]=pc
  float* T1    = (float*)carve((size_t)28 * HO * HS * 4);   // 4.1 MB
  float* UP0   = (float*)carve((size_t)28 * HWO * 4);       // 16.5 MB
  float* UPd   = (float*)carve((size_t)28 * HWO * 4);       // 16.5 MB
  float* UP1   = (float*)carve((size_t)12 * HWO * 4);       // 7.1 MB
  float* UP2   = (float*)carve((size_t)8  * HWO * 4);       // 4.7 MB
  unsigned long long* SORT = (unsigned long long*)carve((size_t)NPAD * 8);  // 8 MB

  float* acc = stats;
  float* gts = stats + 8;
  float* pc  = stats + 32;

  zero_stats_k<<<1, 64, 0, stream>>>(stats);
  build_interp_k<<<(HO + 63) / 64, 64, 0, stream>>>(R, Rt);

  auto gemm = [&](const float* A, const float* Bm, float* D, int M, int N, int K,
                  long sA, long sB, long sD, int slabs) {
    dim3 g(N / 16, M / 32, slabs);
    gemm_wmma_k<<<g, 32, 0, stream>>>(A, Bm, D, M, N, K, sA, sB, sD);
  };

  // preds0 (B*C = 28 slabs): vertical then horizontal resize via WMMA GEMMs
  gemm(R,  preds0, T1,  HO, HS, HS, 0,             (long)HS * HS, (long)HO * HS, 28);
  gemm(T1, Rt,     UP0, HO, WO, HS, (long)HO * HS, 0,             (long)HWO,     28);
  // preds_dsn (28 slabs)
  gemm(R,  predsd, T1,  HO, HS, HS, 0,             (long)HS * HS, (long)HO * HS, 28);
  gemm(T1, Rt,     UPd, HO, WO, HS, (long)HO * HS, 0,             (long)HWO,     28);
  // preds1 (12 slabs)
  gemm(R,  preds1, T1,  HO, HS, HS, 0,             (long)HS * HS, (long)HO * HS, 12);
  gemm(T1, Rt,     UP1, HO, WO, HS, (long)HO * HS, 0,             (long)HWO,     12);
  // preds2 (8 slabs)
  gemm(R,  preds2, T1,  HO, HS, HS, 0,             (long)HS * HS, (long)HO * HS, 8);
  gemm(T1, Rt,     UP2, HO, WO, HS, (long)HO * HS, 0,             (long)HWO,     8);

  const int gridP = NPIX / 256;
  ce_kernel<7><<<gridP, 256, 0, stream>>>(UP0, t0, acc + 0);   // loss_ce (logits)
  ce_kernel<7><<<gridP, 256, 0, stream>>>(UPd, t0, acc + 2);   // loss_dsn

  softmax_inplace_k<7><<<gridP, 256, 0, stream>>>(UP0);
  softmax_inplace_k<3><<<gridP, 256, 0, stream>>>(UP1);
  softmax_inplace_k<2><<<gridP, 256, 0, stream>>>(UP2);

  auto lovasz_branch = [&](const float* probs, const int* lab, int C, int gbase) {
    for (int c = 0; c < C; ++c) {
      build_errors_k<<<NPAD / 256, 256, 0, stream>>>(probs, lab, C, c, SORT,
                                                     gts + (gbase + c));
      for (unsigned int k = 2; k <= NPAD; k <<= 1)
        for (unsigned int j = k >> 1; j > 0; j >>= 1)
          bitonic_step_k<<<NPAD / 256, 256, 0, stream>>>(SORT, j, k);
      lovasz_finalize_k<<<1, 256, 0, stream>>>(SORT, gts, pc, gbase + c);
    }
  };
  lovasz_branch(UP0, t0, 7, 0);   // loss (main lovasz)
  lovasz_branch(UP1, t1, 3, 7);   // loss_hb
  lovasz_branch(UP2, t2, 2, 10);  // loss_fb

  final_combine_k<<<1, 1, 0, stream>>>(acc, gts, pc, (float*)d_out);
}